// SparseGRNLayer_24644522344785
// MI455X (gfx1250) — compile-verified
//
#include <hip/hip_runtime.h>
#include <stdint.h>

#define NGENES     20000
#define NBATCH     128
#define NTFS       1500
#define ACC_STRIDE 33          // pad: conflict-free for [t][lane] hot loop AND lane-varying-t merge
#define TPB        256         // 8 wave32 waves
#define NWAVES     8
#define CHUNK      2048        // edges staged per TDM stage
#define EDGE_BLOCKS 40         // edge chunks across gridDim.x

typedef __attribute__((ext_vector_type(4))) unsigned int v4u;
typedef __attribute__((ext_vector_type(8))) int          v8i;
typedef __attribute__((ext_vector_type(4))) int          v4i;

// ---------------------------------------------------------------------------
// Tensor Data Mover: DMA `ndw` dwords from global `gsrc` into LDS at byte
// offset `lds_byte_off`. D# per CDNA5 ISA:
//   group0: count=1 | lds_addr | global_addr[56:0] | type=2
//   group1: data_size=4B, tensor_dim0=tile_dim0=ndw, tensor_dim1=tile_dim1=1
// Tracked with TENSORcnt (per-wave, in-order); caller waits via s_wait_tensorcnt.
// ---------------------------------------------------------------------------
__device__ __forceinline__ void tdm_load_dwords_to_lds(const void* gsrc,
                                                       uint32_t lds_byte_off,
                                                       uint32_t ndw) {
  uint64_t ga = (uint64_t)(uintptr_t)gsrc;
  v4u g0;
  g0.x = 1u;                                                   // count=1, user D#
  g0.y = lds_byte_off;                                         // lds_addr
  g0.z = (uint32_t)(ga & 0xFFFFFFFFull);                       // global_addr[31:0]
  g0.w = (uint32_t)((ga >> 32) & 0x01FFFFFFull) | (2u << 30);  // addr[56:32] | type=2

  v8i g1;
  g1[0] = (int)(2u << 16);                                   // wg_mask=0, data_size=2 (4B), no flags
  g1[1] = (int)((ndw & 0xFFFFu) << 16);                      // ab_addr=0 | tensor_dim0[15:0]
  g1[2] = (int)(((ndw >> 16) & 0xFFFFu) | (1u << 16));       // tensor_dim0[31:16] | tensor_dim1[15:0]=1
  g1[3] = (int)((ndw & 0xFFFFu) << 16);                      // tensor_dim1[31:16]=0 | tile_dim0=ndw
  g1[4] = 1;                                                 // tile_dim1=1 | tile_dim2=0
  g1[5] = (int)ndw;                                          // tensor_dim0_stride[31:0]
  g1[6] = 0;                                                 // stride0[47:32] | stride1[15:0]
  g1[7] = 0;                                                 // stride1[47:16]

  v4i z4 = {0, 0, 0, 0};
  v8i z8 = {0, 0, 0, 0, 0, 0, 0, 0};
  __builtin_amdgcn_tensor_load_to_lds(g0, g1, z4, z4, z8, 0);
}

// ---------------------------------------------------------------------------
// Main scatter kernel. grid = (EDGE_BLOCKS, NBATCH/32). Each workgroup owns
// 32 batch rows and a contiguous edge range. Full [NTFS x 32] accumulator
// lives in LDS (198 KB of the 320 KB WGP pool); the edge stream is double-
// buffered via TDM async DMA so DMA of stage s+1 overlaps compute of stage s,
// and the stage-0 DMA overlaps the accumulator clear.
// ---------------------------------------------------------------------------
__global__ __launch_bounds__(TPB, 1)
void grn_scatter_kernel(const float* __restrict__ x,
                        const float* __restrict__ xT,
                        const float* __restrict__ wts,
                        const int*   __restrict__ gidx,
                        const int*   __restrict__ tidx,
                        float*       __restrict__ out,
                        int n_edges, int edges_per_block, int use_xt) {
  __shared__ float acc[NTFS * ACC_STRIDE];   // 198,000 B
  __shared__ float s_w[2][CHUNK];            //  16,384 B
  __shared__ int   s_g[2][CHUNK];            //  16,384 B
  __shared__ int   s_t[2][CHUNK];            //  16,384 B

  const int tid   = threadIdx.x;
  const int lane  = tid & 31;               // batch row within block (wave32)
  const int wave  = tid >> 5;               // 0..7
  const int bbase = blockIdx.y * 32;

  const int e0      = blockIdx.x * edges_per_block;
  const int e1      = min(n_edges, e0 + edges_per_block);
  const int nstages = (e1 - e0 + CHUNK - 1) / CHUNK;

  // Prologue FIRST: start DMA of stage 0 so it overlaps the accumulator
  // clear below (disjoint LDS regions).
  if (tid == 0 && nstages > 0) {
    const int n0 = min(CHUNK, e1 - e0);
    tdm_load_dwords_to_lds(wts  + e0, (uint32_t)(uintptr_t)&s_w[0][0], (uint32_t)n0);
    tdm_load_dwords_to_lds(gidx + e0, (uint32_t)(uintptr_t)&s_g[0][0], (uint32_t)n0);
    tdm_load_dwords_to_lds(tidx + e0, (uint32_t)(uintptr_t)&s_t[0][0], (uint32_t)n0);
  }

  for (int i = tid; i < NTFS * ACC_STRIDE; i += TPB) acc[i] = 0.0f;
  __syncthreads();

  for (int s = 0; s < nstages; ++s) {
    const int cur  = s & 1;
    const int base = e0 + s * CHUNK;
    const int n    = min(CHUNK, e1 - base);

    if (tid == 0) {
      if (s + 1 < nstages) {
        // Kick off stage s+1 into the other buffer (last read at stage s-1,
        // released by the trailing barrier of iteration s-1).
        const int nb  = min(CHUNK, e1 - (base + CHUNK));
        const int nxt = cur ^ 1;
        tdm_load_dwords_to_lds(wts  + base + CHUNK, (uint32_t)(uintptr_t)&s_w[nxt][0], (uint32_t)nb);
        tdm_load_dwords_to_lds(gidx + base + CHUNK, (uint32_t)(uintptr_t)&s_g[nxt][0], (uint32_t)nb);
        tdm_load_dwords_to_lds(tidx + base + CHUNK, (uint32_t)(uintptr_t)&s_t[nxt][0], (uint32_t)nb);
        __builtin_amdgcn_s_wait_tensorcnt(3);   // stage s done; s+1 still in flight
      } else {
        __builtin_amdgcn_s_wait_tensorcnt(0);
      }
    }
    __syncthreads();   // stage s data visible to all waves

    // All 32 lanes of a wave process the SAME edge (uniform LDS reads)
    // across 32 batch rows; waves stride through the staged chunk.
#pragma unroll 4
    for (int i = wave; i < n; i += NWAVES) {
      const float we = s_w[cur][i];
      const int   g  = s_g[cur][i];
      const int   t  = s_t[cur][i];
      const float xv = use_xt ? xT[(size_t)g * NBATCH + (bbase + lane)]     // 128B/wave, coalesced
                              : x[(size_t)(bbase + lane) * NGENES + g];     // fallback (L2 resident)
      atomicAdd(&acc[t * ACC_STRIDE + lane], we * xv);   // ds_add_f32, conflict-free
    }
    __syncthreads();   // release buffer `cur` before it is DMA-overwritten at s+2
  }

  // Merge LDS accumulator into global out with coalesced f32 atomics:
  // lanes sweep consecutive t for a fixed batch row (stride-33 -> conflict-free).
  for (int j = tid; j < 32 * NTFS; j += TPB) {
    const int bl = j / NTFS;
    const int t  = j - bl * NTFS;
    atomicAdd(&out[(size_t)(bbase + bl) * NTFS + t], acc[t * ACC_STRIDE + bl]);
  }
}

// ---------------------------------------------------------------------------
__global__ void grn_zero_out_kernel(float* __restrict__ out, int n) {
  int i = blockIdx.x * blockDim.x + threadIdx.x;
  if (i < n) out[i] = 0.0f;
}

// x [128, 20000] row-major -> xT [20000, 128] so per-edge gathers are one
// contiguous 128B line per wave (lane = batch).
__global__ void grn_transpose_kernel(const float* __restrict__ x,
                                     float* __restrict__ xT) {
  __shared__ float tile[32][33];
  const int g0 = blockIdx.x * 32;
  const int b0 = blockIdx.y * 32;
  const int lg = threadIdx.x;          // 0..31
  const int lb = threadIdx.y;          // 0..7
  for (int i = 0; i < 32; i += 8)
    tile[lb + i][lg] = x[(size_t)(b0 + lb + i) * NGENES + (g0 + lg)];
  __syncthreads();
  for (int i = 0; i < 32; i += 8)
    xT[(size_t)(g0 + lb + i) * NBATCH + (b0 + lg)] = tile[lg][lb + i];
}

// ---------------------------------------------------------------------------
extern "C" void kernel_launch(void* const* d_in, const int* in_sizes, int n_in,
                              void* d_out, int out_size, void* d_ws, size_t ws_size,
                              hipStream_t stream) {
  const float* x    = (const float*)d_in[0];   // [128, 20000]
  const float* wts  = (const float*)d_in[1];   // [E]
  const int*   gidx = (const int*)d_in[2];     // [E]
  const int*   tidx = (const int*)d_in[3];     // [E]
  float*       out  = (float*)d_out;           // [128, 1500]
  const int n_edges = in_sizes[1];

  grn_zero_out_kernel<<<dim3((out_size + TPB - 1) / TPB), dim3(TPB), 0, stream>>>(out, out_size);

  float* xT = (float*)d_ws;
  const int use_xt = (ws_size >= (size_t)NGENES * NBATCH * sizeof(float)) ? 1 : 0;
  if (use_xt) {
    grn_transpose_kernel<<<dim3(NGENES / 32, NBATCH / 32), dim3(32, 8), 0, stream>>>(x, xT);
  }

  const int epb = (n_edges + EDGE_BLOCKS - 1) / EDGE_BLOCKS;
  grn_scatter_kernel<<<dim3(EDGE_BLOCKS, NBATCH / 32), dim3(TPB), 0, stream>>>(
      x, xT, wts, gidx, tidx, out, n_edges, epb, use_xt);
}